// SRNet_24842090840116
// MI455X (gfx1250) — compile-verified
//
#include <hip/hip_runtime.h>
#include <cstdint>
#include <cstddef>

// ---------------------------------------------------------------------------
// LUT blob layout (units: floats). All offsets are multiples of 16 floats so
// every 16-wide LUT row is 64B aligned -> ds_load_b128 gathers.
// ---------------------------------------------------------------------------
#define LDW_OFF   0
#define LDW_N     (36*16)
#define HDW_OFF   (LDW_OFF + LDW_N)        // 576
#define HDW_N     (608*16)
#define LPW_OFF   (HDW_OFF + HDW_N)        // 10304
#define LPW_N     (64*16)
#define HPW_OFF   (LPW_OFF + LPW_N)        // 11328
#define HPW_N     (1024*16)
#define X3L_OFF   (HPW_OFF + HPW_N)        // 27712
#define X3L_N     (64*16)
#define X3H_OFF   (X3L_OFF + X3L_N)        // 28736
#define X3H_N     (1024*16)
#define SCL_OFF   (X3H_OFF + X3H_N)        // 45120
#define SCL_N     64
#define BLOB_N    (SCL_OFF + SCL_N)        // 45184 floats = 180736 B < 320KB LDS

static_assert(BLOB_N == 45184, "blob layout");

#ifndef __has_builtin
#define __has_builtin(x) 0
#endif
#if __has_builtin(__builtin_amdgcn_tensor_load_to_lds)
#define HAVE_TDM 1
#else
#define HAVE_TDM 0
#endif

typedef unsigned int v4u __attribute__((ext_vector_type(4)));
typedef int          v8i __attribute__((ext_vector_type(8)));
typedef int          v4i __attribute__((ext_vector_type(4)));

__device__ __forceinline__ float clampf(float v, float lo, float hi) {
    return __builtin_fminf(__builtin_fmaxf(v, lo), hi);
}

// Gather one pre-clipped 16-float LUT row (4 x b128 from LDS) and accumulate.
#define GATHER16(acc, p4) do {                                                \
    float4 _t0 = (p4)[0], _t1 = (p4)[1], _t2 = (p4)[2], _t3 = (p4)[3];        \
    (acc)[0]  += _t0.x; (acc)[1]  += _t0.y; (acc)[2]  += _t0.z; (acc)[3]  += _t0.w; \
    (acc)[4]  += _t1.x; (acc)[5]  += _t1.y; (acc)[6]  += _t1.z; (acc)[7]  += _t1.w; \
    (acc)[8]  += _t2.x; (acc)[9]  += _t2.y; (acc)[10] += _t2.z; (acc)[11] += _t2.w; \
    (acc)[12] += _t3.x; (acc)[13] += _t3.y; (acc)[14] += _t3.z; (acc)[15] += _t3.w; \
} while (0)

// ---------------------------------------------------------------------------
// Kernel 1: pre-clip all LUT tables to [-128,127] into one contiguous blob,
// append the 4x16 per-channel scales. Done once per launch; removes the
// min/max pair from ~2G hot-loop gathers.
// ---------------------------------------------------------------------------
__global__ void srnet_prep(const float* __restrict__ ldw, const float* __restrict__ hdw,
                           const float* __restrict__ lpw, const float* __restrict__ hpw,
                           const float* __restrict__ x3l, const float* __restrict__ x3h,
                           const float* __restrict__ hl1, const float* __restrict__ hh1,
                           const float* __restrict__ hl2, const float* __restrict__ hh2,
                           float* __restrict__ blob) {
    int t = blockIdx.x * blockDim.x + threadIdx.x;
    if (t >= BLOB_N) return;
    float v;
    if (t < HDW_OFF)       v = clampf(ldw[t - LDW_OFF], -128.f, 127.f);
    else if (t < LPW_OFF)  v = clampf(hdw[t - HDW_OFF], -128.f, 127.f);
    else if (t < HPW_OFF)  v = clampf(lpw[t - LPW_OFF], -128.f, 127.f);
    else if (t < X3L_OFF)  v = clampf(hpw[t - HPW_OFF], -128.f, 127.f);
    else if (t < X3H_OFF)  v = clampf(x3l[t - X3L_OFF], -128.f, 127.f);
    else if (t < SCL_OFF)  v = clampf(x3h[t - X3H_OFF], -128.f, 127.f);
    else {
        int s = t - SCL_OFF;   // hl1|hh1|hl2|hh2, 16 each (no clipping)
        v = (s < 16) ? hl1[s] : (s < 32) ? hh1[s - 16]
          : (s < 48) ? hl2[s - 32] : hh2[s - 48];
    }
    blob[t] = v;
}

// ---------------------------------------------------------------------------
// Kernel 2: main LUT pipeline. One thread = one (n,i,j) output pixel:
//   * TDM stages the 180KB LUT blob into LDS (tensor_load_to_lds + tensorcnt)
//   * 9-neighbor depthwise LUT gathers, pointwise LUT x2, all b128 LDS rows
//   * rintf == round-half-even == jnp.round
//   * writes the 4x4 PixelShuffle block as 4 coalesced float4 stores
// ---------------------------------------------------------------------------
__global__ __launch_bounds__(512)
void srnet_main(const int* __restrict__ x, const float* __restrict__ blob,
                float* __restrict__ out, int P) {
    extern __shared__ __align__(16) float sh[];

#if HAVE_TDM
    if (threadIdx.x == 0) {
        // Tensor DMA descriptor: 1-D copy of BLOB_N f32 (tile_dim0=45184 fits 16b)
        unsigned long long ga = (unsigned long long)(uintptr_t)blob;
        unsigned int lds_base = (unsigned int)(uintptr_t)(void*)sh; // low 32b = LDS offset
        const unsigned int N = (unsigned int)BLOB_N;
        v4u g0;
        g0[0] = 1u;                                                   // count=1, no gather
        g0[1] = lds_base;                                             // lds_addr
        g0[2] = (unsigned int)(ga & 0xffffffffull);                   // global_addr lo
        g0[3] = (unsigned int)((ga >> 32) & 0x1ffffffull) | (2u << 30); // ga hi | type=2
        v8i g1;
        g1[0] = (int)(2u << 16);                   // data_size=4B, no mask/pad/iterate
        g1[1] = (int)((N & 0xffffu) << 16);        // tensor_dim0[15:0]   @bit48
        g1[2] = (int)((N >> 16) | (1u << 16));     // dim0[31:16] | tensor_dim1=1
        g1[3] = (int)((N & 0xffffu) << 16);        // tile_dim0 = N       @bit112
        g1[4] = 1;                                 // tile_dim1=1, tile_dim2=0
        g1[5] = (int)N;                            // tensor_dim0_stride lo32
        g1[6] = (int)((N & 0xffffu) << 16);        // stride0 hi=0 | stride1[15:0]
        g1[7] = (int)(N >> 16);                    // stride1[47:16]
        v4i gz  = {0, 0, 0, 0};
        v8i gz8 = {0, 0, 0, 0, 0, 0, 0, 0};
        // 6-arg toolchain form: groups 0..3 + (unused VADDR4-style arg, zero) + cpol
        __builtin_amdgcn_tensor_load_to_lds(g0, g1, gz, gz, gz8, 0);
#if __has_builtin(__builtin_amdgcn_s_wait_tensorcnt)
        __builtin_amdgcn_s_wait_tensorcnt(0);
#else
        asm volatile("s_wait_tensorcnt 0x0" ::: "memory");
#endif
    }
    __syncthreads();
#else
    for (int t = threadIdx.x; t < BLOB_N; t += blockDim.x) sh[t] = blob[t];
    __syncthreads();
#endif

    const float4* __restrict__ sh4 = reinterpret_cast<const float4*>(sh);
    const float*  __restrict__ scl = sh + SCL_OFF;
    const float RCP9 = 1.0f / 9.0f, RCP16 = 0.0625f;

    for (int p = blockIdx.x * blockDim.x + threadIdx.x; p < P; p += gridDim.x * blockDim.x) {
        int n = p / (510 * 510);
        int r = p - n * (510 * 510);
        int i = r / 510;
        int j = r - i * 510;

        const int* xp = x + ((size_t)n * 512 + (size_t)i) * 512 + (size_t)j;
        int vv[9];
        vv[0] = xp[0];    vv[1] = xp[1];    vv[2] = xp[2];
        vv[3] = xp[512];  vv[4] = xp[513];  vv[5] = xp[514];
        vv[6] = xp[1024]; vv[7] = xp[1025]; vv[8] = xp[1026];

        // ---- depthwise 3x3 LUT ----
        float accL[16], accH[16];
        #pragma unroll
        for (int c = 0; c < 16; c++) { accL[c] = 0.f; accH[c] = 0.f; }
        #pragma unroll
        for (int k = 0; k < 9; k++) {
            int xl = vv[k] & 3;                      // idx = x_l + 4k   (0..35)
            int xh = (vv[k] >> 2) + 32 + (k << 6);   // idx = x_h+32+64k (0..607)
            const float4* pl = sh4 + (LDW_OFF / 4) + ((xl + (k << 2)) << 2);
            const float4* ph = sh4 + (HDW_OFF / 4) + (xh << 2);
            GATHER16(accL, pl);
            GATHER16(accH, ph);
        }
        float xlc = (float)(vv[8] & 3);              // x[...,2:,2:] add term
        float xhc = (float)(vv[8] >> 2);

        float oL[16], oH[16];
        #pragma unroll
        for (int c = 0; c < 16; c++) {
            float a = rintf(accL[c] * RCP9) + xlc;
            a = clampf(a, 0.f, 3.f);
            a = rintf(a * scl[c]);                   // hl1
            oL[c] = clampf(a, 0.f, 3.f);
            float b = rintf(accH[c] * RCP9) + xhc;
            b = clampf(b, -32.f, 31.f);
            b = rintf(b * scl[16 + c]);              // hh1
            oH[c] = clampf(b, -32.f, 31.f);
        }

        // ---- pointwise LUT (layer 1) ----
        float pAL[16], pAH[16];
        #pragma unroll
        for (int c = 0; c < 16; c++) { pAL[c] = 0.f; pAH[c] = 0.f; }
        #pragma unroll
        for (int c = 0; c < 16; c++) {
            int il = (c << 2) + (int)oL[c];          // weight_r[c][idx]
            int ih = (c << 6) + (int)oH[c] + 32;     // offset 32
            const float4* pl = sh4 + (LPW_OFF / 4) + (il << 2);
            const float4* ph = sh4 + (HPW_OFF / 4) + (ih << 2);
            GATHER16(pAL, pl);
            GATHER16(pAH, ph);
        }

        float o2L[16], o2H[16];
        #pragma unroll
        for (int c = 0; c < 16; c++) {
            float a = rintf(pAL[c] * RCP16) + oL[c];
            a = clampf(a, 0.f, 3.f);
            a = rintf(a * scl[32 + c]);              // hl2
            o2L[c] = clampf(a, 0.f, 3.f);
            float b = rintf(pAH[c] * RCP16) + oH[c];
            b = clampf(b, -32.f, 31.f);
            b = rintf(b * scl[48 + c]);              // hh2
            o2H[c] = clampf(b, -32.f, 31.f);
        }

        // ---- upsample pointwise LUT ----
        float xAL[16], xAH[16];
        #pragma unroll
        for (int c = 0; c < 16; c++) { xAL[c] = 0.f; xAH[c] = 0.f; }
        #pragma unroll
        for (int c = 0; c < 16; c++) {
            int il = (c << 2) + (int)o2L[c];
            int ih = (c << 6) + (int)o2H[c] + 32;
            const float4* pl = sh4 + (X3L_OFF / 4) + (il << 2);
            const float4* ph = sh4 + (X3H_OFF / 4) + (ih << 2);
            GATHER16(xAL, pl);
            GATHER16(xAH, ph);
        }

        float res[16];
        #pragma unroll
        for (int c = 0; c < 16; c++) {
            float a  = clampf(rintf(xAL[c] * RCP16), -128.f, 127.f);
            float b  = clampf(rintf(xAH[c] * RCP16), -128.f, 127.f);
            float ol = clampf(a + o2L[c], -128.f, 127.f);
            float oh = clampf(b + o2H[c], -128.f, 127.f);
            res[c]   = clampf(oh + ol, -128.f, 127.f);
        }

        // ---- PixelShuffle(4): channel rr*4+ss -> (4i+rr, 4j+ss) ----
        float* op = out + ((size_t)n * 2040 + (size_t)(4 * i)) * 2040 + (size_t)(4 * j);
        #pragma unroll
        for (int rr = 0; rr < 4; rr++) {
            float4 s;
            s.x = res[rr * 4 + 0]; s.y = res[rr * 4 + 1];
            s.z = res[rr * 4 + 2]; s.w = res[rr * 4 + 3];
            *reinterpret_cast<float4*>(op + (size_t)rr * 2040) = s;
        }
    }
}

// ---------------------------------------------------------------------------
extern "C" void kernel_launch(void* const* d_in, const int* in_sizes, int n_in,
                              void* d_out, int out_size, void* d_ws, size_t ws_size,
                              hipStream_t stream) {
    (void)in_sizes; (void)n_in; (void)out_size; (void)ws_size;
    const int*   x   = (const int*)  d_in[0];
    const float* ldw = (const float*)d_in[1];
    const float* hdw = (const float*)d_in[2];
    const float* lpw = (const float*)d_in[3];
    const float* hpw = (const float*)d_in[4];
    const float* x3l = (const float*)d_in[5];
    const float* x3h = (const float*)d_in[6];
    const float* hl1 = (const float*)d_in[7];
    const float* hh1 = (const float*)d_in[8];
    const float* hl2 = (const float*)d_in[9];
    const float* hh2 = (const float*)d_in[10];
    float* blob = (float*)d_ws;
    float* out  = (float*)d_out;

    srnet_prep<<<(BLOB_N + 255) / 256, 256, 0, stream>>>(
        ldw, hdw, lpw, hpw, x3l, x3h, hl1, hh1, hl2, hh2, blob);

    const int P = 6 * 510 * 510;                    // 1,560,600 output pixels
    const int threads = 512;                        // 16 waves; LDS caps 1 WG/WGP
    const int blocks  = (P + threads - 1) / threads;
    srnet_main<<<blocks, threads, BLOB_N * sizeof(float), stream>>>(x, blob, out, P);
}